// GIN_60086592471619
// MI455X (gfx1250) — compile-verified
//
#include <hip/hip_runtime.h>
#include <hip/hip_bf16.h>
#include <math.h>

typedef float v2f __attribute__((ext_vector_type(2)));
typedef float v8f __attribute__((ext_vector_type(8)));

#define WID 128
#define LDSW 132          // padded row stride (conflict-free: 132 % 64 = 4)

__device__ __forceinline__ float gelu_erf(float v) {
    return 0.5f * v * (1.0f + erff(v * 0.70710678118654752440f));
}

// ---------------- foot: h = gelu(x @ W_foot^T + b_foot) ----------------
__global__ void foot_kernel(const float* __restrict__ x,
                            const float* __restrict__ Wf,
                            const float* __restrict__ bf,
                            float* __restrict__ h, int n_nodes) {
    int t = blockIdx.x * blockDim.x + threadIdx.x;
    if (t >= n_nodes * WID) return;
    int node = t >> 7;
    int f = t & (WID - 1);
    const float* xr = x + (size_t)node * 16;
    const float* wr = Wf + (size_t)f * 16;
    float s = bf[f];
#pragma unroll
    for (int k = 0; k < 16; ++k) s += xr[k] * wr[k];
    h[t] = gelu_erf(s);
}

// ---------------- scatter: agg[col[e]] += h[row[e]] ----------------
__global__ void scatter_kernel(const float* __restrict__ h,
                               const int* __restrict__ row,
                               const int* __restrict__ col,
                               float* __restrict__ agg, int n_edges) {
    int t = blockIdx.x * blockDim.x + threadIdx.x;
    int e = t >> 5;
    if (e >= n_edges) return;
    int c4 = (t & 31) << 2;
    int rs = row[e];
    int cd = col[e];
    const float4 v = *(const float4*)(h + (size_t)rs * WID + c4);
    float* dst = agg + (size_t)cd * WID + c4;
    atomicAdd(dst + 0, v.x);
    atomicAdd(dst + 1, v.y);
    atomicAdd(dst + 2, v.z);
    atomicAdd(dst + 3, v.w);
}

// ---------------- fused GIN MLP block (fp32 WMMA 16x16x4) ----------------
// z = h + agg; t = gelu(z @ W1^T + b1); h = h + t @ W2^T + b2   (in place)
// 8 waves/block, each wave owns 32 nodes (two 16-row A tiles sharing every
// B fragment -> half the LDS traffic per WMMA).
__global__ __launch_bounds__(256) void mlp_wmma_kernel(
    float* __restrict__ h, const float* __restrict__ agg,
    const float* __restrict__ W1, const float* __restrict__ b1,
    const float* __restrict__ W2, const float* __restrict__ b2, int n_nodes) {
    extern __shared__ float smem[];
    float* sW = smem;                    // 128 rows x 132 stride (weights, padded)
    float* sT = smem + WID * LDSW;       // 8 waves x 32 rows x 132 stride

    const int tid  = threadIdx.x;
    const int wave = tid >> 5;
    const int lane = tid & 31;
    const int half = lane >> 4;          // 0: lanes 0-15, 1: lanes 16-31
    const int l16  = lane & 15;

    // stage W1 -> LDS (padded rows)
    for (int i = tid; i < WID * WID; i += 256) {
        int f = i >> 7, k = i & (WID - 1);
        sW[f * LDSW + k] = W1[i];
    }

    const int base = blockIdx.x * 256 + wave * 32;  // first node of this wave

    // hoist per-lane biases (feat = nt*16 + l16)
    float bias1[8], bias2[8];
#pragma unroll
    for (int nt = 0; nt < 8; ++nt) {
        bias1[nt] = b1[nt * 16 + l16];
        bias2[nt] = b2[nt * 16 + l16];
    }

    // ---- load A fragments of Z = h + agg for two 16-row tiles ----
    // A layout (16x4 f32): lane(l16)=row M; .x holds K=kbase+2*half, .y = +1
    int row0 = base + l16;        if (row0 >= n_nodes) row0 = n_nodes - 1;
    int row1 = base + 16 + l16;   if (row1 >= n_nodes) row1 = n_nodes - 1;
    const float* h0 = h   + (size_t)row0 * WID;
    const float* g0 = agg + (size_t)row0 * WID;
    const float* h1 = h   + (size_t)row1 * WID;
    const float* g1 = agg + (size_t)row1 * WID;
    v2f a0[32], a1[32];
#pragma unroll
    for (int ks = 0; ks < 32; ++ks) {
        int k0 = ks * 4 + 2 * half;
        a0[ks].x = h0[k0]     + g0[k0];
        a0[ks].y = h0[k0 + 1] + g0[k0 + 1];
        a1[ks].x = h1[k0]     + g1[k0];
        a1[ks].y = h1[k0 + 1] + g1[k0 + 1];
    }
    __syncthreads();   // W1 staged

    // ---- GEMM1: T = gelu(Z @ W1^T + b1) -> LDS (row-major per wave) ----
    float* Tw = sT + wave * (32 * LDSW);
    for (int nt = 0; nt < 8; ++nt) {
        int feat = nt * 16 + l16;
        v8f c0, c1;
#pragma unroll
        for (int r = 0; r < 8; ++r) { c0[r] = bias1[nt]; c1[r] = bias1[nt]; }
#pragma unroll
        for (int ks = 0; ks < 32; ++ks) {
            int k0 = ks * 4 + 2 * half;
            v2f bf;                                  // B[k][n] = W1[n][k]
            bf.x = sW[feat * LDSW + k0];
            bf.y = sW[feat * LDSW + k0 + 1];
            c0 = __builtin_amdgcn_wmma_f32_16x16x4_f32(
                     false, a0[ks], false, bf, (short)0, c0, false, false);
            c1 = __builtin_amdgcn_wmma_f32_16x16x4_f32(
                     false, a1[ks], false, bf, (short)0, c1, false, false);
        }
        // C layout: VGPR r, lane -> (M = r + 8*half, N = l16); GELU, park in LDS
#pragma unroll
        for (int r = 0; r < 8; ++r) {
            int m = r + 8 * half;
            Tw[m * LDSW + feat]        = gelu_erf(c0[r]);
            Tw[(m + 16) * LDSW + feat] = gelu_erf(c1[r]);
        }
    }

    // re-stage W2 over W1 (all waves must be done reading sW)
    __syncthreads();
    for (int i = tid; i < WID * WID; i += 256) {
        int f = i >> 7, k = i & (WID - 1);
        sW[f * LDSW + k] = W2[i];
    }
    __syncthreads();

    // ---- reload T as A fragments (LDS relayout C->A) ----
    v2f at0[32], at1[32];
#pragma unroll
    for (int ks = 0; ks < 32; ++ks) {
        int k0 = ks * 4 + 2 * half;
        at0[ks].x = Tw[l16 * LDSW + k0];
        at0[ks].y = Tw[l16 * LDSW + k0 + 1];
        at1[ks].x = Tw[(16 + l16) * LDSW + k0];
        at1[ks].y = Tw[(16 + l16) * LDSW + k0 + 1];
    }

    // ---- GEMM2: h = h_old + T @ W2^T + b2 (residual folded into C init) ----
    for (int nt = 0; nt < 8; ++nt) {
        int feat = nt * 16 + l16;
        v8f c0, c1;
#pragma unroll
        for (int r = 0; r < 8; ++r) {
            int n0 = base + r + 8 * half;
            int n1 = n0 + 16;
            int c0i = n0 < n_nodes ? n0 : n_nodes - 1;
            int c1i = n1 < n_nodes ? n1 : n_nodes - 1;
            c0[r] = bias2[nt] + h[(size_t)c0i * WID + feat];
            c1[r] = bias2[nt] + h[(size_t)c1i * WID + feat];
        }
#pragma unroll
        for (int ks = 0; ks < 32; ++ks) {
            int k0 = ks * 4 + 2 * half;
            v2f bf;
            bf.x = sW[feat * LDSW + k0];
            bf.y = sW[feat * LDSW + k0 + 1];
            c0 = __builtin_amdgcn_wmma_f32_16x16x4_f32(
                     false, at0[ks], false, bf, (short)0, c0, false, false);
            c1 = __builtin_amdgcn_wmma_f32_16x16x4_f32(
                     false, at1[ks], false, bf, (short)0, c1, false, false);
        }
#pragma unroll
        for (int r = 0; r < 8; ++r) {
            int n0 = base + r + 8 * half;
            int n1 = n0 + 16;
            if (n0 < n_nodes) h[(size_t)n0 * WID + feat] = c0[r];
            if (n1 < n_nodes) h[(size_t)n1 * WID + feat] = c1[r];
        }
    }
}

// ---------------- head: only the B selected nodes ----------------
__global__ void head_kernel(const float* __restrict__ h,
                            const float* __restrict__ Wh,
                            const float* __restrict__ bh,
                            const int* __restrict__ center,
                            const int* __restrict__ ptr,
                            float* __restrict__ out, int nB, int nO) {
    int t = blockIdx.x * blockDim.x + threadIdx.x;
    if (t >= nB * nO) return;
    int g = t / nO;
    int o = t - g * nO;
    int node = center[g] + ptr[g];
    const float* hr = h + (size_t)node * WID;
    const float* wr = Wh + (size_t)o * WID;
    float s = bh[o];
    for (int k = 0; k < WID; ++k) s += gelu_erf(hr[k]) * wr[k];
    out[t] = s;
}

extern "C" void kernel_launch(void* const* d_in, const int* in_sizes, int n_in,
                              void* d_out, int out_size, void* d_ws, size_t ws_size,
                              hipStream_t stream) {
    const float* x      = (const float*)d_in[0];
    const int*   ei     = (const int*)  d_in[1];
    const int*   center = (const int*)  d_in[2];
    const int*   ptr    = (const int*)  d_in[3];
    const float* Wf     = (const float*)d_in[4];
    const float* bf     = (const float*)d_in[5];
    const float* W1     = (const float*)d_in[6];
    const float* b1     = (const float*)d_in[7];
    const float* W2     = (const float*)d_in[8];
    const float* b2     = (const float*)d_in[9];
    const float* Wh     = (const float*)d_in[10];
    const float* bh     = (const float*)d_in[11];
    float* out = (float*)d_out;

    const int N    = in_sizes[0] / 16;    // 100000
    const int E    = in_sizes[1] / 2;     // 600000
    const int Bg   = in_sizes[2];         // 100
    const int NBLK = in_sizes[7] / WID;   // 6
    const int nO   = out_size / Bg;       // 7

    float* h   = (float*)d_ws;
    float* agg = h + (size_t)N * WID;

    // 128*132 (weights) + 8 waves * 32 rows * 132 (T scratch) floats
    const size_t lds_bytes = (size_t)(WID * LDSW + 8 * 32 * LDSW) * sizeof(float);

    foot_kernel<<<(N * WID + 255) / 256, 256, 0, stream>>>(x, Wf, bf, h, N);

    for (int l = 0; l < NBLK; ++l) {
        hipMemsetAsync(agg, 0, (size_t)N * WID * sizeof(float), stream);
        scatter_kernel<<<(E * 32 + 255) / 256, 256, 0, stream>>>(h, ei, ei + E, agg, E);
        mlp_wmma_kernel<<<(N + 255) / 256, 256, lds_bytes, stream>>>(
            h, agg,
            W1 + (size_t)l * WID * WID, b1 + (size_t)l * WID,
            W2 + (size_t)l * WID * WID, b2 + (size_t)l * WID, N);
    }

    head_kernel<<<(Bg * nO + 127) / 128, 128, 0, stream>>>(h, Wh, bh, center, ptr, out, Bg, nO);
}